// FinalCompressedTokens_35785667510440
// MI455X (gfx1250) — compile-verified
//
#include <hip/hip_runtime.h>
#include <hip/hip_bf16.h>

// Problem constants (fixed by setup_inputs)
constexpr int B    = 2;
constexpr int T    = 4096;   // T_cmp
constexpr int C    = 4096;
constexpr int H    = 32;
constexpr int KVH  = 8;      // groups = 4
constexpr int D    = 128;
constexpr int W    = 512;
constexpr int KSEL = 1024;   // top-k
constexpr int TOUT = T + KSEL; // 5120

typedef float v2f __attribute__((ext_vector_type(2)));
typedef float v8f __attribute__((ext_vector_type(8)));

// ---------------------------------------------------------------------------
// Kernel 0: zero the importance accumulator
// ---------------------------------------------------------------------------
__global__ void zero_kernel(float* __restrict__ p, int n) {
    int i = blockIdx.x * blockDim.x + threadIdx.x;
    if (i < n) p[i] = 0.0f;
}

// ---------------------------------------------------------------------------
// Kernel 1: fused scores-GEMM + softmax + importance accumulation.
// Grid: B*H*(W/32) = 1024 WGs, 256 threads (8 waves). Each WG owns a
// 32(q) x 4096(k) score tile; each wave owns a 512-wide N strip.
// Pass 1: WMMA + exp -> per-row Z. Pass 2: recompute WMMA tile, accumulate
// exp(s)/Z per column into importance via global fp32 atomics.
// ---------------------------------------------------------------------------
__global__ __launch_bounds__(256) void importance_kernel(
    const float* __restrict__ q_w,     // (B,H,W,D)
    const float* __restrict__ km_cmp,  // (B,KVH,T,D)
    float* __restrict__ imp)           // (B,T)
{
    const int wg  = blockIdx.x;
    const int qt  = wg & 15;          // 16 q-tiles of 32 rows
    const int h   = (wg >> 4) & 31;
    const int b   = wg >> 9;
    const int m0  = qt * 32;

    const int lane = threadIdx.x & 31;
    const int wave = threadIdx.x >> 5;
    const int lo   = lane & 15;
    const int hi   = lane >> 4;

    __shared__ float rowsum[32];
    if (threadIdx.x < 32) rowsum[threadIdx.x] = 0.0f;
    __syncthreads();

    const float scale = 0.08838834764831845f;  // D^-0.5

    // ---- load A (32 q-rows x 128) into VGPRs, WMMA A layout ----
    const float* qbase = q_w + ((size_t)(b * H + h) * W) * D;
    const v2f* arow0 = (const v2f*)(qbase + (size_t)(m0 + lo) * D);
    const v2f* arow1 = (const v2f*)(qbase + (size_t)(m0 + 16 + lo) * D);
    v2f a0[32], a1[32];
#pragma unroll
    for (int g = 0; g < 32; ++g) {
        a0[g] = arow0[2 * g + hi];
        a1[g] = arow1[2 * g + hi];
    }

    const float* kb = km_cmp + ((size_t)(b * KVH + (h >> 2)) * T) * D;
    const int n_wave = wave * 512;

    // ---------------- pass 1: row sums of exp(s) ----------------
    float rs0[8], rs1[8];
#pragma unroll
    for (int i = 0; i < 8; ++i) { rs0[i] = 0.0f; rs1[i] = 0.0f; }

    for (int nt = 0; nt < 32; ++nt) {
        const int n0 = n_wave + nt * 16;
        const v2f* brow = (const v2f*)(kb + (size_t)(n0 + lo) * D);
        v8f c0 = {0.f,0.f,0.f,0.f,0.f,0.f,0.f,0.f};
        v8f c1 = {0.f,0.f,0.f,0.f,0.f,0.f,0.f,0.f};
#pragma unroll
        for (int g = 0; g < 32; ++g) {
            v2f bv = brow[2 * g + hi];
            c0 = __builtin_amdgcn_wmma_f32_16x16x4_f32(false, a0[g], false, bv,
                                                       (short)0, c0, false, false);
            c1 = __builtin_amdgcn_wmma_f32_16x16x4_f32(false, a1[g], false, bv,
                                                       (short)0, c1, false, false);
        }
#pragma unroll
        for (int i = 0; i < 8; ++i) {
            rs0[i] += __expf(c0[i] * scale);
            rs1[i] += __expf(c1[i] * scale);
        }
    }
#pragma unroll
    for (int i = 0; i < 8; ++i) {
        atomicAdd(&rowsum[i + hi * 8], rs0[i]);        // rows m0 + (i + 8*hi)
        atomicAdd(&rowsum[16 + i + hi * 8], rs1[i]);   // rows m0+16 + (i + 8*hi)
    }
    __syncthreads();

    float iz0[8], iz1[8];
#pragma unroll
    for (int i = 0; i < 8; ++i) {
        iz0[i] = 1.0f / rowsum[i + hi * 8];
        iz1[i] = 1.0f / rowsum[16 + i + hi * 8];
    }

    // ---------------- pass 2: column accumulation ----------------
    for (int nt = 0; nt < 32; ++nt) {
        const int n0 = n_wave + nt * 16;
        const v2f* brow = (const v2f*)(kb + (size_t)(n0 + lo) * D);
        v8f c0 = {0.f,0.f,0.f,0.f,0.f,0.f,0.f,0.f};
        v8f c1 = {0.f,0.f,0.f,0.f,0.f,0.f,0.f,0.f};
#pragma unroll
        for (int g = 0; g < 32; ++g) {
            v2f bv = brow[2 * g + hi];
            c0 = __builtin_amdgcn_wmma_f32_16x16x4_f32(false, a0[g], false, bv,
                                                       (short)0, c0, false, false);
            c1 = __builtin_amdgcn_wmma_f32_16x16x4_f32(false, a1[g], false, bv,
                                                       (short)0, c1, false, false);
        }
        float col = 0.0f;
#pragma unroll
        for (int i = 0; i < 8; ++i) {
            col += __expf(c0[i] * scale) * iz0[i];
            col += __expf(c1[i] * scale) * iz1[i];
        }
        col += __shfl_xor(col, 16);    // combine the two lane halves (same column)
        if (lane < 16)
            atomicAdd(&imp[(size_t)b * T + n0 + lane], col);
    }
}

// ---------------------------------------------------------------------------
// Kernel 2: per-batch top-1024 (bitonic sort of packed (key, ~idx) u64 in LDS,
// tie-break smaller-index-first to match jax.lax.top_k), then expand-mask and
// exclusive prefix count -> start positions.
// Grid: B blocks of 1024 threads.
// ---------------------------------------------------------------------------
__global__ __launch_bounds__(1024) void topk_kernel(
    const float* __restrict__ imp,
    int* __restrict__ gmask, int* __restrict__ gstart)
{
    const int b   = blockIdx.x;
    const int tid = threadIdx.x;

    __shared__ unsigned long long keys[4096];
    __shared__ unsigned int smask[4096];
    __shared__ unsigned int part[1024];

    for (int i = tid; i < T; i += 1024) {
        unsigned u   = __float_as_uint(imp[(size_t)b * T + i]);
        unsigned key = (u & 0x80000000u) ? ~u : (u | 0x80000000u); // order-preserving
        keys[i]  = ((unsigned long long)key << 32) | (unsigned)(~i);
        smask[i] = 0u;
    }
    __syncthreads();

    // ascending bitonic sort; top-k ends up at indices [T-KSEL, T)
    for (int k = 2; k <= 4096; k <<= 1) {
        for (int j = k >> 1; j > 0; j >>= 1) {
            for (int i = tid; i < 4096; i += 1024) {
                int ixj = i ^ j;
                if (ixj > i) {
                    bool up = ((i & k) == 0);
                    unsigned long long x = keys[i], y = keys[ixj];
                    if (up == (x > y)) { keys[i] = y; keys[ixj] = x; }
                }
            }
            __syncthreads();
        }
    }

    {
        unsigned low = (unsigned)(keys[(T - KSEL) + tid] & 0xFFFFFFFFull);
        unsigned t   = ~low;           // original token index
        smask[t] = 1u;
    }
    __syncthreads();

    // prefix count of masked tokens (4 elements per thread + Hillis-Steele)
    unsigned l0 = smask[4 * tid + 0], l1 = smask[4 * tid + 1];
    unsigned l2 = smask[4 * tid + 2], l3 = smask[4 * tid + 3];
    unsigned lsum = l0 + l1 + l2 + l3;
    part[tid] = lsum;
    __syncthreads();
    unsigned x = lsum;
    for (int off = 1; off < 1024; off <<= 1) {
        unsigned y = (tid >= off) ? part[tid - off] : 0u;
        __syncthreads();
        x += y;
        part[tid] = x;
        __syncthreads();
    }
    unsigned excl = x - lsum;          // masked count before this chunk
    unsigned c0 = excl, c1 = c0 + l0, c2 = c1 + l1, c3 = c2 + l2;

    const int base = b * T + 4 * tid;
    gmask[base + 0] = (int)l0;  gstart[base + 0] = 4 * tid + 0 + (int)c0;
    gmask[base + 1] = (int)l1;  gstart[base + 1] = 4 * tid + 1 + (int)c1;
    gmask[base + 2] = (int)l2;  gstart[base + 2] = 4 * tid + 2 + (int)c2;
    gmask[base + 3] = (int)l3;  gstart[base + 3] = 4 * tid + 3 + (int)c3;
}

// ---------------------------------------------------------------------------
// Kernel 3: row scatter. One 256-thread WG per token; float4 (b128) copies.
// Output coverage is total (4096 + 1024 = 5120 rows) -> no zero-init needed.
// ---------------------------------------------------------------------------
__global__ __launch_bounds__(256) void scatter_kernel(
    const float* __restrict__ x_m,     // (B, 2T, C)
    const float* __restrict__ xm_cmp,  // (B, T, C)
    const int* __restrict__ gmask,
    const int* __restrict__ gstart,
    float* __restrict__ out)           // (B, TOUT, C)
{
    const int blk = blockIdx.x;
    const int b = blk >> 12;           // / T
    const int t = blk & (T - 1);       // % T
    const int m  = gmask[b * T + t];
    const int st = gstart[b * T + t];

    const float* s1 = m ? (x_m    + ((size_t)b * 2 * T + 2 * t) * C)
                        : (xm_cmp + ((size_t)b * T + t) * C);
    const float4* src1 = (const float4*)s1;
    float4* dst1 = (float4*)(out + ((size_t)b * TOUT + st) * C);
    for (int i = threadIdx.x; i < C / 4; i += 256) dst1[i] = src1[i];

    if (m) {
        const float4* src2 = (const float4*)(x_m + ((size_t)b * 2 * T + 2 * t + 1) * C);
        float4* dst2 = dst1 + C / 4;
        for (int i = threadIdx.x; i < C / 4; i += 256) dst2[i] = src2[i];
    }
}

// ---------------------------------------------------------------------------
extern "C" void kernel_launch(void* const* d_in, const int* in_sizes, int n_in,
                              void* d_out, int out_size, void* d_ws, size_t ws_size,
                              hipStream_t stream) {
    (void)in_sizes; (void)n_in; (void)out_size; (void)ws_size;
    const float* x_m    = (const float*)d_in[0];
    const float* xm_cmp = (const float*)d_in[1];
    const float* q_w    = (const float*)d_in[2];
    const float* km_cmp = (const float*)d_in[3];
    float* out = (float*)d_out;

    float* imp   = (float*)d_ws;                         // B*T floats
    int*   gmask = (int*)((char*)d_ws + (size_t)B * T * sizeof(float));
    int*   gstart = gmask + B * T;

    zero_kernel<<<(B * T + 255) / 256, 256, 0, stream>>>(imp, B * T);
    importance_kernel<<<B * H * (W / 32), 256, 0, stream>>>(q_w, km_cmp, imp);
    topk_kernel<<<B, 1024, 0, stream>>>(imp, gmask, gstart);
    scatter_kernel<<<B * T, 256, 0, stream>>>(x_m, xm_cmp, gmask, gstart, out);
}